// EuclideanFarthestPointPrePooling_57921928954501
// MI455X (gfx1250) — compile-verified
//
#include <hip/hip_runtime.h>
#include <hip/hip_bf16.h>

// ---------------------------------------------------------------------------
// EuclideanFarthestPointPrePooling for MI455X (gfx1250, wave32)
//
// B=8, C=256, H=W=112 -> n=12544 points per image, m=256 selected, 255 steps.
// Per step: d = norms + norms[last] - 2*(V@y); dist=min(dist,d); argmax(dist).
// L2-resident streaming problem (102.8MB input < 192MB L2). Matvec done with
// V_WMMA_F32_16X16X4_F32 (full f32 precision -> argmax fidelity vs reference).
// ---------------------------------------------------------------------------

#define BATCH 8
#define CCH   256
#define HW    112
#define NPTS  (HW * HW)      // 12544 = 392 * 32
#define MSEL  256
#define STEPS (MSEL - 1)     // 255
#define GWG   14             // workgroups per batch
#define WAVES 14             // waves per workgroup (448 threads)
#define BLOCK (WAVES * 32)
// total waves per batch = GWG*WAVES = 196; groups of 32 points = 392 = 196*2

typedef __attribute__((ext_vector_type(2))) float v2f;
typedef __attribute__((ext_vector_type(8))) float v8f;

// ---------------------------------------------------------------------------
// helpers
// ---------------------------------------------------------------------------
__device__ __forceinline__ float sel8(v8f a, int r) {
  // runtime select of one of 8 accumulator VGPRs -> chain of v_cndmask
  float v = a[0];
  v = (r == 1) ? a[1] : v;
  v = (r == 2) ? a[2] : v;
  v = (r == 3) ? a[3] : v;
  v = (r == 4) ? a[4] : v;
  v = (r == 5) ? a[5] : v;
  v = (r == 6) ? a[6] : v;
  v = (r == 7) ? a[7] : v;
  return v;
}

__device__ __forceinline__ unsigned long long pack_key(float v, int idx) {
  // orderable float key; ties broken toward SMALLER index (jnp.argmax rule)
  unsigned int b = __float_as_uint(v);
  b ^= (b >> 31) ? 0xFFFFFFFFu : 0x80000000u;
  return ((unsigned long long)b << 32) |
         (unsigned long long)(0xFFFFFFFFu - (unsigned int)idx);
}

// Dot of every point in a 32-point group against y (staged in LDS), using two
// 16x16x4 f32 WMMA accumulation chains (tile0 = rows 0..15, tile1 = 16..31).
// Returns this lane's point dot per the C/D layout:
//   lanes 0-7  -> tile0 rows 0-7   (acc0[lane&7])
//   lanes 8-15 -> tile1 rows 0-7   (acc1[lane&7])
//   lanes 16-23-> tile0 rows 8-15  (acc0[lane&7])
//   lanes 24-31-> tile1 rows 8-15  (acc1[lane&7])
__device__ __forceinline__ float group_dot(const float* __restrict__ xb,
                                           const float* __restrict__ ylds,
                                           int pbase, int lane, int dk) {
  v8f acc0 = {0.f, 0.f, 0.f, 0.f, 0.f, 0.f, 0.f, 0.f};
  v8f acc1 = {0.f, 0.f, 0.f, 0.f, 0.f, 0.f, 0.f, 0.f};
  const int m = lane & 15;
  const float* pa0 = xb + pbase + m;        // A tile0: point column pbase+m
  const float* pa1 = xb + pbase + 16 + m;   // A tile1
#pragma unroll 8
  for (int t = 0; t < CCH / 4; ++t) {
    const int k = t * 4 + dk;               // this lane's K pair (ISA A layout)
    v2f a0, a1, bv;
    a0.x = pa0[(size_t)k * NPTS];
    a0.y = pa0[(size_t)(k + 1) * NPTS];
    a1.x = pa1[(size_t)k * NPTS];
    a1.y = pa1[(size_t)(k + 1) * NPTS];
    bv.x = ylds[k];                         // y broadcast to all 16 B columns
    bv.y = ylds[k + 1];
    acc0 = __builtin_amdgcn_wmma_f32_16x16x4_f32(false, a0, false, bv,
                                                 (short)0, acc0, false, false);
    acc1 = __builtin_amdgcn_wmma_f32_16x16x4_f32(false, a1, false, bv,
                                                 (short)0, acc1, false, false);
  }
  const int r = lane & 7;
  const float d0 = sel8(acc0, r);
  const float d1 = sel8(acc1, r);
  return ((lane >> 3) & 1) ? d1 : d0;
}

// ---------------------------------------------------------------------------
// kernels
// ---------------------------------------------------------------------------
__global__ void fps_init_kernel(unsigned long long* __restrict__ slots,
                                unsigned int* __restrict__ counters,
                                int* __restrict__ selected) {
  int i = blockIdx.x * blockDim.x + threadIdx.x;
  if (i < BATCH * STEPS) {
    slots[i] = 0ull;
    counters[i] = 0u;
  }
  if (i < BATCH) selected[i * MSEL] = 0;  // idx[0] = 0
}

__global__ void fps_norms_kernel(const float* __restrict__ x,
                                 float* __restrict__ norms) {
  int i = blockIdx.x * blockDim.x + threadIdx.x;
  if (i >= BATCH * NPTS) return;
  int b = i / NPTS, p = i - b * NPTS;
  const float* xb = x + (size_t)b * CCH * NPTS + p;
  float s = 0.f;
#pragma unroll 8
  for (int c = 0; c < CCH; ++c) {
    float v = xb[(size_t)c * NPTS];
    s = fmaf(v, v, s);
  }
  norms[i] = s;
}

__global__ __launch_bounds__(BLOCK) void fps_main_kernel(
    const float* __restrict__ x, const float* __restrict__ norms,
    int* __restrict__ selected, unsigned long long* __restrict__ slots,
    unsigned int* __restrict__ counters) {
  const int b = blockIdx.x / GWG;
  const int g = blockIdx.x % GWG;
  const int tid = threadIdx.x;
  const int w = tid >> 5;
  const int lane = tid & 31;

  const float* xb = x + (size_t)b * CCH * NPTS;
  const float* nb = norms + (size_t)b * NPTS;

  __shared__ float ylds[CCH];
  __shared__ float nlast_s;
  __shared__ unsigned long long red[WAVES];
  __shared__ int next_idx_s;

  // static ownership: this wave handles 2 groups of 32 points for all steps
  const int wv = g * WAVES + w;                  // 0..195
  const int gb0 = (wv * 2) * 32;
  const int gb1 = (wv * 2 + 1) * 32;
  const int tile = (lane >> 3) & 1;
  const int row = (lane & 7) + ((lane >= 16) ? 8 : 0);
  const int p0 = gb0 + tile * 16 + row;          // this lane's point, group 0
  const int p1 = gb1 + tile * 16 + row;          // this lane's point, group 1
  const int dk = (lane < 16) ? 0 : 2;            // K pair offset (ISA layout)

  const float norm0 = nb[p0];
  const float norm1 = nb[p1];
  float dist0 = __builtin_huge_valf();
  float dist1 = __builtin_huge_valf();
  int last = 0;

  for (int s = 0; s < STEPS; ++s) {
    // stage y = V[last] (gather one channel row of x) + its norm into LDS
    for (int c = tid; c < CCH; c += BLOCK) ylds[c] = xb[(size_t)c * NPTS + last];
    if (tid == 0) nlast_s = nb[last];
    __syncthreads();
    const float nlast = nlast_s;

    // d = norms_p + norm_last - 2*dot  (WMMA f32 matvec), dist = min(dist, d)
    const float dot0 = group_dot(xb, ylds, gb0, lane, dk);
    const float dot1 = group_dot(xb, ylds, gb1, lane, dk);
    dist0 = fminf(dist0, fmaf(-2.f, dot0, norm0 + nlast));
    dist1 = fminf(dist1, fmaf(-2.f, dot1, norm1 + nlast));

    // lane-local argmax (strict > keeps smaller index on ties; p0 < p1)
    float bv = dist0;
    int bi = p0;
    if (dist1 > bv) { bv = dist1; bi = p1; }
    unsigned long long pk = pack_key(bv, bi);

    // wave32 reduction
#pragma unroll
    for (int off = 16; off > 0; off >>= 1) {
      unsigned long long o = __shfl_xor(pk, off, 32);
      pk = (o > pk) ? o : pk;
    }
    if (lane == 0) red[w] = pk;
    __syncthreads();

    if (tid == 0) {
      unsigned long long mx = red[0];
#pragma unroll
      for (int i = 1; i < WAVES; ++i) mx = (red[i] > mx) ? red[i] : mx;
      unsigned long long* slot = &slots[b * STEPS + s];
      unsigned int* cnt = &counters[b * STEPS + s];
      __hip_atomic_fetch_max(slot, mx, __ATOMIC_RELEASE,
                             __HIP_MEMORY_SCOPE_AGENT);
      // step-indexed arrival barrier across the 14 WGs of this batch
      unsigned int arrived = __hip_atomic_fetch_add(
                                 cnt, 1u, __ATOMIC_ACQ_REL,
                                 __HIP_MEMORY_SCOPE_AGENT) + 1u;
      while (arrived < GWG) {
        __builtin_amdgcn_s_sleep(2);
        arrived = __hip_atomic_load(cnt, __ATOMIC_ACQUIRE,
                                    __HIP_MEMORY_SCOPE_AGENT);
      }
      unsigned long long win = __hip_atomic_load(slot, __ATOMIC_RELAXED,
                                                 __HIP_MEMORY_SCOPE_AGENT);
      int idx = (int)(0xFFFFFFFFu - (unsigned int)(win & 0xFFFFFFFFull));
      next_idx_s = idx;
      if (g == 0) selected[b * MSEL + s + 1] = idx;
    }
    __syncthreads();
    last = next_idx_s;
  }
}

__global__ void fps_gather_kernel(const float* __restrict__ x,
                                  const int* __restrict__ selected,
                                  float* __restrict__ out) {
  int i = blockIdx.x * blockDim.x + threadIdx.x;
  if (i >= BATCH * CCH * MSEL) return;
  int j = i % MSEL;
  int bc = i / MSEL;          // b*C + c
  int b = bc / CCH;
  int pt = selected[b * MSEL + j];
  out[i] = x[(size_t)bc * NPTS + pt];   // out[b][c][j] = x[b][c][pt]
}

// ---------------------------------------------------------------------------
// launch
// ---------------------------------------------------------------------------
extern "C" void kernel_launch(void* const* d_in, const int* in_sizes, int n_in,
                              void* d_out, int out_size, void* d_ws,
                              size_t ws_size, hipStream_t stream) {
  (void)in_sizes; (void)n_in; (void)out_size; (void)ws_size;
  const float* x = (const float*)d_in[0];
  float* out = (float*)d_out;

  // workspace layout (all 8-byte aligned)
  char* ws = (char*)d_ws;
  int* selected = (int*)ws;                                   // B*MSEL i32
  size_t off = (size_t)BATCH * MSEL * sizeof(int);
  unsigned long long* slots = (unsigned long long*)(ws + off); // B*STEPS u64
  off += (size_t)BATCH * STEPS * sizeof(unsigned long long);
  unsigned int* counters = (unsigned int*)(ws + off);          // B*STEPS u32
  off += (size_t)BATCH * STEPS * sizeof(unsigned int);
  off = (off + 7) & ~(size_t)7;
  float* norms = (float*)(ws + off);                           // B*NPTS f32

  {
    int n = BATCH * STEPS;
    fps_init_kernel<<<(n + 255) / 256, 256, 0, stream>>>(slots, counters,
                                                         selected);
  }
  {
    int n = BATCH * NPTS;
    fps_norms_kernel<<<(n + 255) / 256, 256, 0, stream>>>(x, norms);
  }
  fps_main_kernel<<<BATCH * GWG, BLOCK, 0, stream>>>(x, norms, selected, slots,
                                                     counters);
  {
    int n = BATCH * CCH * MSEL;
    fps_gather_kernel<<<(n + 255) / 256, 256, 0, stream>>>(x, selected, out);
  }
}